// NoiseLayer_84232898609373
// MI455X (gfx1250) — compile-verified
//
#include <hip/hip_runtime.h>
#include <stdint.h>

// ---------------------------------------------------------------------------
// NoiseLayer: flip each bit of each f32 with p = 0.01 (BER), deterministically.
//
// Memory-bound streaming kernel (134 MB @ 23.3 TB/s ~ 5.8 us floor).
// Data path per lane (all inside one asm block so LLVM can't elide the LDS
// round-trip and ordering/waits are explicit):
//   global_load_async_to_lds_b128   (CDNA5 async copy, ASYNCcnt)
//   s_wait_asynccnt 0
//   ds_load_b128                    (DScnt)
//   s_wait_dscnt 0
// then VALU bit-flips and a non-temporal b128 global store.
//
// RNG: geometric skip sampling over each 128-bit chunk:
//   skip = floor(log2(u) / log2(1-p)), u ~ U(0,1]
// with a stateless counter-hash PRNG keyed by chunk index -> deterministic.
// ~2.3 draws + 1 v_log_f32 (TRANS) per 16 bytes -> compute far under roofline.
// ---------------------------------------------------------------------------

typedef uint32_t u32x4 __attribute__((ext_vector_type(4)));

static __device__ __forceinline__ uint32_t mix32(uint32_t x) {
  x ^= x >> 16; x *= 0x7feb352du;
  x ^= x >> 15; x *= 0x846ca68bu;
  x ^= x >> 16;
  return x;
}

struct Rng {
  uint32_t s;
  __device__ __forceinline__ uint32_t next() {
    s = s * 747796405u + 2891336453u;   // LCG advance
    return mix32(s);                    // output whitening
  }
};

static __device__ __forceinline__ float log2_fast(float x) {
#if __has_builtin(__builtin_amdgcn_logf)
  return __builtin_amdgcn_logf(x);      // v_log_f32 (TRANS; co-executes w/ VALU)
#else
  return __log2f(x);
#endif
}

// Flip bits of a 128-bit chunk {lo, hi}; each bit flips independently w/ p=0.01.
static __device__ __forceinline__ void flip_bits_128(uint64_t& lo, uint64_t& hi,
                                                     uint32_t seed) {
  Rng rng{mix32(seed ^ 0x9e3779b9u)};
  const float kInvLog2Q = -68.96756f;   // 1 / log2(0.99)
  int pos = -1;
  for (;;) {
    uint32_t r = rng.next() | 1u;                       // u in (0, 1]
    float u = (float)r * 2.3283064365386963e-10f;       // * 2^-32
    int skip = (int)(log2_fast(u) * kInvLog2Q);         // geometric, >= 0
    pos += 1 + skip;
    if (pos >= 128) break;
    uint64_t b = 1ull << (pos & 63);                    // v_lshlrev_b64
    if (pos < 64) lo ^= b; else hi ^= b;                // cndmask'd xors
  }
}

__global__ __launch_bounds__(256) void noise_bitflip_kernel(
    const uint32_t* __restrict__ in_w, uint32_t* __restrict__ out_w,
    uint32_t n4, uint32_t n_words) {
  __shared__ u32x4 stage[256];          // only LDS object -> byte offset 0

  const uint32_t tid = threadIdx.x;
  const uint32_t gid = blockIdx.x * 256u + tid;

  if (gid < n4) {
    uint32_t lds_off = tid * 16u;       // LDS byte address for this lane
    uint32_t voff    = gid * 16u;       // byte offset from SGPR base (GVS mode)

    // Fused, explicitly-ordered CDNA5 async staging sequence. The unused %4
    // operand escapes `stage` so the LDS frame is kept allocated. Output is
    // early-clobber: it is written after all inputs are consumed.
    u32x4 w;
    asm volatile(
        "global_load_async_to_lds_b128 %1, %2, %3\n\t"
        "s_wait_asynccnt 0x0\n\t"
        "ds_load_b128 %0, %1\n\t"
        "s_wait_dscnt 0x0"
        : "=&v"(w)
        : "v"(lds_off), "v"(voff), "s"(in_w), "v"((const void*)stage)
        : "memory");

    uint64_t lo = ((uint64_t)w.y << 32) | (uint64_t)w.x;
    uint64_t hi = ((uint64_t)w.w << 32) | (uint64_t)w.z;
    flip_bits_128(lo, hi, gid);
    w.x = (uint32_t)lo; w.y = (uint32_t)(lo >> 32);
    w.z = (uint32_t)hi; w.w = (uint32_t)(hi >> 32);

    // Dead stream: bypass cache retention with a non-temporal b128 store.
    __builtin_nontemporal_store(w, (u32x4*)out_w + gid);
  }

  // Scalar tail (n_words not multiple of 4); inert for the 2048x8192 case.
  if (gid == 0) {
    for (uint32_t i = n4 * 4u; i < n_words; ++i) {
      uint64_t lo = (uint64_t)in_w[i];
      uint64_t hi = 0;
      flip_bits_128(lo, hi, 0x5bd1e995u + i);
      out_w[i] = (uint32_t)lo;
    }
  }
}

extern "C" void kernel_launch(void* const* d_in, const int* in_sizes, int n_in,
                              void* d_out, int out_size, void* d_ws, size_t ws_size,
                              hipStream_t stream) {
  (void)in_sizes; (void)n_in; (void)d_ws; (void)ws_size;
  const uint32_t n_words = (uint32_t)out_size;          // 2048*8192 f32 words
  const uint32_t n4      = n_words >> 2;                // 128-bit chunks
  const uint32_t* in_w   = (const uint32_t*)d_in[0];
  uint32_t*       out_w  = (uint32_t*)d_out;

  uint32_t blocks = (n4 + 255u) / 256u;
  if (blocks == 0) blocks = 1;                          // keep tail path alive
  noise_bitflip_kernel<<<blocks, 256, 0, stream>>>(in_w, out_w, n4, n_words);
}